// SymmetricContraction_26585847562987
// MI455X (gfx1250) — compile-verified
//
#include <hip/hip_runtime.h>

typedef float v4f __attribute__((ext_vector_type(4)));

static constexpr int kMul    = 128;
static constexpr int kPaths  = 5;
static constexpr int kElems  = 10;
static constexpr int kRowV4  = (4 * kMul) / 4;             // 128 float4 per x row
static constexpr int kOutV4  = kMul / 4;                   // 32 float4 per out row
static constexpr int kWTotV4 = kElems * kPaths * (kMul/4); // 1600 float4 = 25.6 KB

#define INV_SQRT3F 0.57735026918962576f

__global__ __launch_bounds__(256)
void symcon_kernel(const float* __restrict__ x,
                   const float* __restrict__ weight,
                   const int*   __restrict__ indices,
                   float*       __restrict__ out,
                   int nrows, int rows_per_wave)
{
    // Cache the full (10,5,128) weight table in LDS once per block using the
    // CDNA5 async global->LDS datapath (no VGPR staging, ASYNCcnt-tracked).
    __shared__ v4f wlds[kWTotV4];

    {
        const v4f* wg = (const v4f*)weight;
        unsigned lds_base = (unsigned)(uintptr_t)(void*)&wlds[0]; // low 32b = LDS offset
        for (int i = threadIdx.x; i < kWTotV4; i += 256) {
            unsigned    lds_off = lds_base + (unsigned)i * 16u;
            const v4f*  gptr    = wg + i;
            asm volatile("global_load_async_to_lds_b128 %0, %1, off"
                         :: "v"(lds_off), "v"(gptr)
                         : "memory");
        }
#if __has_builtin(__builtin_amdgcn_s_wait_asynccnt)
        __builtin_amdgcn_s_wait_asynccnt(0);
#else
        asm volatile("s_wait_asynccnt 0x0" ::: "memory");
#endif
    }
    __syncthreads();

    const int lane   = threadIdx.x & 31;
    const int waveId = (blockIdx.x << 3) | (threadIdx.x >> 5);   // 8 waves / block

    long long row0 = (long long)waveId * rows_per_wave;
    long long row1 = row0 + rows_per_wave;
    if (row1 > nrows) row1 = nrows;

    const v4f* x4 = (const v4f*)x;
    v4f*       o4 = (v4f*)out;

    for (long long row = row0; row < row1; ++row) {
        const v4f* xr = x4 + row * kRowV4;

        // Keep the linear HBM stream ahead of us (global_prefetch_b8).
        __builtin_prefetch(xr + kRowV4 + lane, 0, 0);

        // Index is wave-uniform (one row per wave) -> force to SGPR.
        int e = __builtin_amdgcn_readfirstlane(indices[row]);

        // Streaming, zero-reuse data: non-temporal b128 loads (1 GB stream
        // vs 192 MB L2 -> NT avoids thrashing the weight table out of cache).
        v4f s  = __builtin_nontemporal_load(xr       + lane);   // x[:, 0:128]
        v4f v0 = __builtin_nontemporal_load(xr + 32  + lane);   // x[:, 128:256]
        v4f v1 = __builtin_nontemporal_load(xr + 64  + lane);   // x[:, 256:384]
        v4f v2 = __builtin_nontemporal_load(xr + 96  + lane);   // x[:, 384:512]

        const v4f* wl = wlds + e * (kPaths * kOutV4);
        v4f w0 = wl[0 * kOutV4 + lane];
        v4f w1 = wl[1 * kOutV4 + lane];
        v4f w2 = wl[2 * kOutV4 + lane];
        v4f w3 = wl[3 * kOutV4 + lane];
        v4f w4 = wl[4 * kOutV4 + lane];

        v4f vv = (v0 * v0 + v1 * v1 + v2 * v2) * INV_SQRT3F;
        v4f s2 = s * s;
        v4f r  = w0 * s + w1 * s2 + w2 * vv + w3 * (s2 * s) + w4 * (s * vv);

        __builtin_nontemporal_store(r, o4 + row * kOutV4 + lane);
    }
}

extern "C" void kernel_launch(void* const* d_in, const int* in_sizes, int n_in,
                              void* d_out, int out_size, void* d_ws, size_t ws_size,
                              hipStream_t stream)
{
    const float* x       = (const float*)d_in[0];
    const float* weight  = (const float*)d_in[1];
    const int*   indices = (const int*)d_in[2];
    float*       out     = (float*)d_out;

    const int B = in_sizes[2];            // indices has one entry per row

    // Persistent-ish grid: enough waves to hide HBM latency, few enough that
    // the 25.6 KB LDS weight preload (per block) is amortized over ~31 rows/wave.
    const int blocks     = 2048;          // x 8 waves = 16384 waves
    const int totalWaves = blocks * 8;
    const int rpw        = (B + totalWaves - 1) / totalWaves;

    symcon_kernel<<<blocks, 256, 0, stream>>>(x, weight, indices, out, B, rpw);
}